// DC_and_CE_loss_67791763800111
// MI455X (gfx1250) — compile-verified
//
#include <hip/hip_runtime.h>
#include <hip/hip_bf16.h>
#include <math.h>

typedef __attribute__((ext_vector_type(2))) float v2f;
typedef __attribute__((ext_vector_type(8))) float v8f;

#define SDIM 128
#define S3   (1u << 21)      // 128^3
#define NVOX (1u << 22)      // 2 * 128^3
#define CF   16
#define RR   10
#define NBINS 4096
#define TOPN 250
#define SMOOTHV 1e-5f

// ---- ws float layout ----
#define WS_POSSUM 0      // 16 floats
#define WS_POSCNT 16
#define WS_STDN   17     // 16 floats
#define WS_CE     33
#define WS_TP     34     // 3
#define WS_FP     37     // 3
#define WS_FN     40     // 3
#define WS_PLSUM  43
#define WS_MISSUM 44
#define WS_MISCNT 45
#define WS_NEGSUM 46
#define WS_NEGCNT 47
#define WS_THRESH 48
#define WS_HIST   64     // 4096 floats
#define WS_COS    8192   // NVOX floats
#define WS_MASKA_OFF ((size_t)(WS_COS + NVOX) * 4)
#define WS_MASKB_OFF (WS_MASKA_OFF + (size_t)NVOX)

__device__ __forceinline__ float waveSum(float v) {
  #pragma unroll
  for (int off = 16; off > 0; off >>= 1) v += __shfl_xor(v, off, 32);
  return v;
}

// ---- 0: zero scalar accumulators + histogram ----
__global__ void k_zero(float* ws) {
  int i = blockIdx.x * blockDim.x + threadIdx.x;
  if (i < WS_HIST + NBINS) ws[i] = 0.f;
}

// ---- 1: sum feature vectors over positive voxels ----
__global__ void k_possum(const float* __restrict__ feat, const int* __restrict__ tgt,
                         float* ws) {
  unsigned tid = blockIdx.x * blockDim.x + threadIdx.x;
  unsigned stride = gridDim.x * blockDim.x;
  float acc[CF];
  #pragma unroll
  for (int c = 0; c < CF; c++) acc[c] = 0.f;
  float cnt = 0.f;
  for (unsigned v = tid; v < NVOX; v += stride) {
    if (tgt[v] == 1) {
      cnt += 1.f;
      unsigned b = v >> 21, s = v & (S3 - 1);
      const float* fb = feat + (size_t)b * CF * S3 + s;
      #pragma unroll
      for (int c = 0; c < CF; c++) acc[c] += fb[(size_t)c * S3];
    }
  }
  #pragma unroll
  for (int c = 0; c < CF; c++) acc[c] = waveSum(acc[c]);
  cnt = waveSum(cnt);
  if ((threadIdx.x & 31) == 0) {
    #pragma unroll
    for (int c = 0; c < CF; c++) atomicAdd(&ws[WS_POSSUM + c], acc[c]);
    atomicAdd(&ws[WS_POSCNT], cnt);
  }
}

// ---- 2: normalize the mean positive feature (single wave) ----
__global__ void k_stdn(float* ws) {
  int l = threadIdx.x;
  float cnt = ws[WS_POSCNT];
  float m = 0.f;
  if (l < CF) m = ws[WS_POSSUM + l] / fmaxf(cnt, 1.f);
  if (cnt <= 0.f) m = 0.f;
  float nsq = waveSum(m * m);
  float n = fmaxf(sqrtf(nsq), 1e-12f);
  if (l < CF) ws[WS_STDN + l] = m / n;
}

// ---- 3: cosine map via v_wmma_f32_16x16x4_f32 ----
// Wave processes 16 voxels/group. B(4x16) = feature channels, A(16x4) = std_n
// replicated across rows => D[r][n] = dot(f_n, std_n) for every r.
__global__ void __launch_bounds__(256) k_cos(const float* __restrict__ feat,
                                             const int* __restrict__ tgt,
                                             float* ws) {
  __shared__ float lhist[NBINS];
  for (int i = threadIdx.x; i < NBINS; i += blockDim.x) lhist[i] = 0.f;
  __syncthreads();

  const float* stdn = ws + WS_STDN;
  float* cosmap = ws + WS_COS;
  int lane = threadIdx.x & 31;
  int half = lane >> 4;      // which half-wave supplies K pair {2h,2h+1}
  int l16  = lane & 15;      // column / voxel-in-group

  v2f a[4];
  #pragma unroll
  for (int s = 0; s < 4; s++) {
    int ch = 4 * s + 2 * half;
    a[s].x = stdn[ch];
    a[s].y = stdn[ch + 1];
  }

  unsigned nGroups = NVOX / 16;
  unsigned wid = blockIdx.x * (blockDim.x >> 5) + (threadIdx.x >> 5);
  unsigned nW  = gridDim.x * (blockDim.x >> 5);
  float plSum = 0.f;

  for (unsigned g = wid; g < nGroups; g += nW) {
    unsigned v = g * 16u + (unsigned)l16;
    unsigned b = v >> 21, sidx = v & (S3 - 1);
    const float* fb = feat + (size_t)b * CF * S3 + sidx;
    v8f acc = {};
    float nl = 0.f;
    #pragma unroll
    for (int s = 0; s < 4; s++) {
      int ch = 4 * s + 2 * half;
      float b0 = fb[(size_t)ch * S3];
      float b1 = fb[(size_t)(ch + 1) * S3];
      nl += b0 * b0 + b1 * b1;
      v2f bb; bb.x = b0; bb.y = b1;
      acc = __builtin_amdgcn_wmma_f32_16x16x4_f32(false, a[s], false, bb,
                                                  (short)0, acc, false, false);
    }
    float dot = acc[0];                       // replicated over rows/cols
    float nf = nl + __shfl_xor(nl, 16, 32);   // combine channel halves
    float c  = dot / fmaxf(sqrtf(nf), 1e-12f);
    if (half == 0) {
      cosmap[v] = c;
      int t = tgt[v];
      if (t == 1) {
        plSum += 1.f - c;
      } else if (t == 0) {
        float x = (c + 1.f) * (float)(NBINS / 2);
        int bin = (int)x;
        bin = bin < 0 ? 0 : (bin > NBINS - 1 ? NBINS - 1 : bin);
        atomicAdd(&lhist[bin], 1.f);
      }
    }
  }
  plSum = waveSum(plSum);
  if (lane == 0) atomicAdd(&ws[WS_PLSUM], plSum);
  __syncthreads();
  for (int i = threadIdx.x; i < NBINS; i += blockDim.x) {
    float hv = lhist[i];
    if (hv != 0.f) atomicAdd(&ws[WS_HIST + i], hv);
  }
}

// ---- 4: CE + dice partial sums, and positive byte mask ----
__global__ void k_cedice(const float* __restrict__ net, const int* __restrict__ tgt,
                         float* ws, unsigned char* __restrict__ maskP) {
  unsigned tid = blockIdx.x * blockDim.x + threadIdx.x;
  unsigned stride = gridDim.x * blockDim.x;
  float ce = 0.f, tp0 = 0.f, tp1 = 0.f, tp2 = 0.f;
  float fp0 = 0.f, fp1 = 0.f, fp2 = 0.f, fn0 = 0.f, fn1 = 0.f, fn2 = 0.f;
  for (unsigned v = tid; v < NVOX; v += stride) {
    int t = tgt[v];
    unsigned b = v >> 21, s = v & (S3 - 1);
    const float* nb = net + (size_t)b * 3 * S3 + s;
    float x0 = nb[0], x1 = nb[(size_t)S3], x2 = nb[(size_t)2 * S3];
    float m = fmaxf(x0, fmaxf(x1, x2));
    float e0 = __expf(x0 - m), e1 = __expf(x1 - m), e2 = __expf(x2 - m);
    float sum = e0 + e1 + e2;
    float lse = m + __logf(sum);
    float xt = (t == 0) ? x0 : ((t == 1) ? x1 : x2);
    ce += lse - xt;
    float inv = 1.f / sum;
    float p0 = e0 * inv, p1 = e1 * inv, p2 = e2 * inv;
    float y0 = (t == 0) ? 1.f : 0.f;
    float y1 = (t == 1) ? 1.f : 0.f;
    float y2 = (t == 2) ? 1.f : 0.f;
    tp0 += p0 * y0; fp0 += p0 * (1.f - y0); fn0 += (1.f - p0) * y0;
    tp1 += p1 * y1; fp1 += p1 * (1.f - y1); fn1 += (1.f - p1) * y1;
    tp2 += p2 * y2; fp2 += p2 * (1.f - y2); fn2 += (1.f - p2) * y2;
    maskP[v] = (t == 1) ? 1 : 0;
  }
  ce = waveSum(ce);
  tp0 = waveSum(tp0); tp1 = waveSum(tp1); tp2 = waveSum(tp2);
  fp0 = waveSum(fp0); fp1 = waveSum(fp1); fp2 = waveSum(fp2);
  fn0 = waveSum(fn0); fn1 = waveSum(fn1); fn2 = waveSum(fn2);
  if ((threadIdx.x & 31) == 0) {
    atomicAdd(&ws[WS_CE], ce);
    atomicAdd(&ws[WS_TP + 0], tp0); atomicAdd(&ws[WS_TP + 1], tp1); atomicAdd(&ws[WS_TP + 2], tp2);
    atomicAdd(&ws[WS_FP + 0], fp0); atomicAdd(&ws[WS_FP + 1], fp1); atomicAdd(&ws[WS_FP + 2], fp2);
    atomicAdd(&ws[WS_FN + 0], fn0); atomicAdd(&ws[WS_FN + 1], fn1); atomicAdd(&ws[WS_FN + 2], fn2);
  }
}

// ---- 5: scan histogram from top for the ~250th largest negative cosine ----
__global__ void k_scan(float* ws) {
  if (blockIdx.x == 0 && threadIdx.x == 0) {
    float cum = 0.f;
    float th = -1e30f;
    for (int i = NBINS - 1; i >= 0; i--) {
      cum += ws[WS_HIST + i];
      if (cum >= (float)TOPN) { th = -1.f + 2.f * (float)i / (float)NBINS; break; }
    }
    ws[WS_THRESH] = th;
  }
}

// ---- 6: build hardest-negative byte mask ----
__global__ void k_himask(const int* __restrict__ tgt, const float* __restrict__ ws,
                         unsigned char* __restrict__ maskM) {
  unsigned v = blockIdx.x * blockDim.x + threadIdx.x;
  if (v >= NVOX) return;
  float th = ws[WS_THRESH];
  maskM[v] = (tgt[v] == 0 && ws[WS_COS + v] >= th) ? 1 : 0;
}

// ---- 7: separable binary box dilation (window 21) along one axis ----
__global__ void k_dilate(const unsigned char* __restrict__ src,
                         unsigned char* __restrict__ dst, unsigned stride) {
  unsigned v = blockIdx.x * blockDim.x + threadIdx.x;
  if (v >= NVOX) return;
  int c = (int)((v / stride) & 127u);
  int lo = c > RR ? c - RR : 0;
  int hi = c + RR < SDIM ? c + RR : SDIM - 1;
  unsigned base = v - (unsigned)c * stride;
  unsigned char m = 0;
  for (int cc = lo; cc <= hi; cc++) m |= src[base + (unsigned)cc * stride];
  dst[v] = m;
}

// ---- 8: masked relu-mean accumulation over (dilated & ~pos) ----
__global__ void k_maskmean(const unsigned char* __restrict__ dil,
                           const int* __restrict__ tgt, const float* __restrict__ ws,
                           float* sumP, float* cntP) {
  unsigned tid = blockIdx.x * blockDim.x + threadIdx.x;
  unsigned stride = gridDim.x * blockDim.x;
  float s = 0.f, c = 0.f;
  for (unsigned v = tid; v < NVOX; v += stride) {
    if (dil[v] && tgt[v] != 1) {
      float cv = ws[WS_COS + v];
      s += cv > 0.f ? cv : 0.f;
      c += 1.f;
    }
  }
  s = waveSum(s); c = waveSum(c);
  if ((threadIdx.x & 31) == 0) { atomicAdd(sumP, s); atomicAdd(cntP, c); }
}

// ---- 9: combine everything into the scalar loss ----
__global__ void k_final(const float* ws, float* out) {
  if (blockIdx.x == 0 && threadIdx.x == 0) {
    float N = (float)NVOX;
    float ce = ws[WS_CE] / N;
    float dc = 0.f;
    #pragma unroll
    for (int c = 1; c < 3; c++) {
      float tp = ws[WS_TP + c], fp = ws[WS_FP + c], fn = ws[WS_FN + c];
      dc += (2.f * tp + SMOOTHV) / fmaxf(2.f * tp + fp + fn + SMOOTHV, 1e-8f);
    }
    float dcLoss = -dc * 0.5f;
    float cp = ws[WS_POSCNT];
    float pl = cp > 0.f ? ws[WS_PLSUM] / fmaxf(cp, 1.f) : 0.f;
    float mc = ws[WS_MISCNT];
    float ml = mc > 0.f ? ws[WS_MISSUM] / fmaxf(mc, 1.f) : 0.f;
    float nc = ws[WS_NEGCNT];
    float nl = nc > 0.f ? ws[WS_NEGSUM] / fmaxf(nc, 1.f) : 0.f;
    out[0] = ce + dcLoss + 5.f * (pl + ml + nl);
  }
}

extern "C" void kernel_launch(void* const* d_in, const int* in_sizes, int n_in,
                              void* d_out, int out_size, void* d_ws, size_t ws_size,
                              hipStream_t stream) {
  const float* feat = (const float*)d_in[0];
  const float* net  = (const float*)d_in[1];
  const int*   tgt  = (const int*)d_in[2];
  float* ws = (float*)d_ws;
  unsigned char* mA = (unsigned char*)d_ws + WS_MASKA_OFF;
  unsigned char* mB = (unsigned char*)d_ws + WS_MASKB_OFF;
  float* out = (float*)d_out;

  k_zero<<<(WS_HIST + NBINS + 255) / 256, 256, 0, stream>>>(ws);
  k_possum<<<2048, 256, 0, stream>>>(feat, tgt, ws);
  k_stdn<<<1, 32, 0, stream>>>(ws);
  k_cos<<<1024, 256, 0, stream>>>(feat, tgt, ws);
  k_cedice<<<2048, 256, 0, stream>>>(net, tgt, ws, mA);
  k_scan<<<1, 32, 0, stream>>>(ws);

  dim3 dil((NVOX + 255) / 256);
  // dilate positive mask: W, H, D -> result in mB
  k_dilate<<<dil, 256, 0, stream>>>(mA, mB, 1u);
  k_dilate<<<dil, 256, 0, stream>>>(mB, mA, 128u);
  k_dilate<<<dil, 256, 0, stream>>>(mA, mB, 16384u);
  k_maskmean<<<2048, 256, 0, stream>>>(mB, tgt, ws, &ws[WS_MISSUM], &ws[WS_MISCNT]);

  // hardest-negative mask, dilate, accumulate
  k_himask<<<dil, 256, 0, stream>>>(tgt, ws, mA);
  k_dilate<<<dil, 256, 0, stream>>>(mA, mB, 1u);
  k_dilate<<<dil, 256, 0, stream>>>(mB, mA, 128u);
  k_dilate<<<dil, 256, 0, stream>>>(mA, mB, 16384u);
  k_maskmean<<<2048, 256, 0, stream>>>(mB, tgt, ws, &ws[WS_NEGSUM], &ws[WS_NEGCNT]);

  k_final<<<1, 1, 0, stream>>>(ws, out);
}